// GCNModel_75557064671960
// MI455X (gfx1250) — compile-verified
//
#include <hip/hip_runtime.h>
#include <hip/hip_bf16.h>
#include <stddef.h>

// ---------------------------------------------------------------------------
// GCN: out = spmm(A, spmm(A, spmm(A, x) @ W1 + b1) @ W2 + b2) @ W3 + b3
// N = 50000, E = 800000, dims 128 -> 128 -> 128 -> 64, all fp32.
//
// SpMM: wave32 per edge; lane l moves float4 at dim 4*l (coalesced 512B/edge),
//       scatter via global_atomic_add_f32 (25.6MB target lives in 192MB L2).
// GEMM: V_WMMA_F32_16X16X4_F32 (native fp32 WMMA). One wave computes a
//       16 x (NT*16) output stripe, reusing each A fragment across NT WMMAs.
// ---------------------------------------------------------------------------

typedef __attribute__((ext_vector_type(2))) float v2f;
typedef __attribute__((ext_vector_type(8))) float v8f;

// ---------------- zero-fill (accumulator init before scatter-add) ----------
__global__ __launch_bounds__(256) void zero_f4_kernel(float4* __restrict__ p,
                                                      size_t n4) {
  size_t i = (size_t)blockIdx.x * blockDim.x + threadIdx.x;
  size_t stride = (size_t)gridDim.x * blockDim.x;
  for (; i < n4; i += stride) p[i] = make_float4(0.f, 0.f, 0.f, 0.f);
}

// ---------------- SpMM: y[dst] += vals[e] * x[src], D = 128 ----------------
__global__ __launch_bounds__(256) void spmm_kernel(
    const int* __restrict__ dstIdx, const int* __restrict__ srcIdx,
    const float* __restrict__ vals, const float* __restrict__ x,
    float* __restrict__ y, int E) {
  const int lane = threadIdx.x & 31;
  int wave = (int)((blockIdx.x * blockDim.x + threadIdx.x) >> 5);
  const int nWaves = (int)((gridDim.x * blockDim.x) >> 5);
  for (int e = wave; e < E; e += nWaves) {
    const int s = srcIdx[e];
    const int d = dstIdx[e];
    const float v = vals[e];
    const float4 xv =
        *(const float4*)(x + (size_t)s * 128 + (size_t)lane * 4);
    float* yp = y + (size_t)d * 128 + (size_t)lane * 4;
    atomicAdd(yp + 0, xv.x * v);
    atomicAdd(yp + 1, xv.y * v);
    atomicAdd(yp + 2, xv.z * v);
    atomicAdd(yp + 3, xv.w * v);
  }
}

// ---------------- Dense GEMM + bias via fp32 WMMA --------------------------
// out[N x NT*16] = S[N x 128] @ W[128 x NT*16] + bias
// One wave per 16-row tile; loops k in steps of 4 (V_WMMA_F32_16X16X4_F32).
// A frag layout: lanes 0-15 -> M=lane, K={k,k+1}; lanes 16-31 -> K={k+2,k+3}.
// B frag layout mirrors A (lanes = N columns). C/D: VGPR j -> M=j (+8 for
// lanes 16-31), N = lane&15.
template <int NT>
__global__ __launch_bounds__(128) void gemm_bias_kernel(
    const float* __restrict__ S, const float* __restrict__ W,
    const float* __restrict__ bias, float* __restrict__ out, int nRowTiles) {
  const int lane = threadIdx.x & 31;
  const int waveInBlock = threadIdx.x >> 5;
  const int rowTile = blockIdx.x * (blockDim.x >> 5) + waveInBlock;
  if (rowTile >= nRowTiles) return;  // wave-uniform: EXEC stays all-ones

  const int hi = lane >> 4;  // 0: K pair {k,k+1} / M 0..7 ; 1: {k+2,k+3} / M 8..15
  const int lo = lane & 15;
  const int ldw = NT * 16;
  const float* Arow = S + (size_t)(rowTile * 16 + lo) * 128;

  v8f acc[NT] = {};

#pragma unroll
  for (int k = 0; k < 128; k += 4) {
    const int kk = k + hi * 2;
    v2f a = *(const v2f*)(Arow + kk);  // 8B-aligned (kk even)
#pragma unroll
    for (int t = 0; t < NT; ++t) {
      const int col = t * 16 + lo;
      v2f b;
      b.x = W[(size_t)kk * ldw + col];
      b.y = W[(size_t)(kk + 1) * ldw + col];
      acc[t] = __builtin_amdgcn_wmma_f32_16x16x4_f32(
          /*neg_a=*/false, a, /*neg_b=*/false, b,
          /*c_mod=*/(short)0, acc[t], /*reuse_a=*/false, /*reuse_b=*/false);
    }
  }

#pragma unroll
  for (int t = 0; t < NT; ++t) {
    const int col = t * 16 + lo;
    const float bv = bias[col];
#pragma unroll
    for (int j = 0; j < 8; ++j) {
      const int m = j + hi * 8;
      out[(size_t)(rowTile * 16 + m) * ldw + col] = acc[t][j] + bv;
    }
  }
}

// ---------------------------------------------------------------------------
extern "C" void kernel_launch(void* const* d_in, const int* in_sizes, int n_in,
                              void* d_out, int out_size, void* d_ws,
                              size_t ws_size, hipStream_t stream) {
  const float* x = (const float*)d_in[0];     // [N, 128]
  const int* adj = (const int*)d_in[1];       // [2, E] (row0 = dst, row1 = src)
  const float* vals = (const float*)d_in[2];  // [E]
  const float* W1 = (const float*)d_in[3];    // [128, 128]
  const float* b1 = (const float*)d_in[4];
  const float* W2 = (const float*)d_in[5];    // [128, 128]
  const float* b2 = (const float*)d_in[6];
  const float* W3 = (const float*)d_in[7];    // [128, 64]
  const float* b3 = (const float*)d_in[8];
  float* out = (float*)d_out;                 // [N, 64]

  const int E = in_sizes[2];
  const int N = in_sizes[0] / 128;
  const int* dstIdx = adj;
  const int* srcIdx = adj + E;

  float* ws0 = (float*)d_ws;                  // spmm accumulator [N,128]
  float* ws1 = ws0 + (size_t)N * 128;         // dense layer output [N,128]

  const size_t n4 = (size_t)N * 128 / 4;
  const int zeroBlocks = 2048;
  const int spmmBlocks = (int)(((long long)E * 32 + 255) / 256);
  const int nRowTiles = N / 16;               // 50000 / 16 = 3125 exact
  const int gemmBlocks = (nRowTiles + 3) / 4; // 4 waves per block

  // ---- layer 1: s = spmm(x); h = s @ W1 + b1
  zero_f4_kernel<<<zeroBlocks, 256, 0, stream>>>((float4*)ws0, n4);
  spmm_kernel<<<spmmBlocks, 256, 0, stream>>>(dstIdx, srcIdx, vals, x, ws0, E);
  gemm_bias_kernel<8><<<gemmBlocks, 128, 0, stream>>>(ws0, W1, b1, ws1,
                                                      nRowTiles);

  // ---- layer 2: s = spmm(h); h = s @ W2 + b2
  zero_f4_kernel<<<zeroBlocks, 256, 0, stream>>>((float4*)ws0, n4);
  spmm_kernel<<<spmmBlocks, 256, 0, stream>>>(dstIdx, srcIdx, vals, ws1, ws0, E);
  gemm_bias_kernel<8><<<gemmBlocks, 128, 0, stream>>>(ws0, W2, b2, ws1,
                                                      nRowTiles);

  // ---- layer 3: s = spmm(h); out = s @ W3 + b3
  zero_f4_kernel<<<zeroBlocks, 256, 0, stream>>>((float4*)ws0, n4);
  spmm_kernel<<<spmmBlocks, 256, 0, stream>>>(dstIdx, srcIdx, vals, ws1, ws0, E);
  gemm_bias_kernel<4><<<gemmBlocks, 128, 0, stream>>>(ws0, W3, b3, out,
                                                      nRowTiles);
}